// HeteroGNN_25709674234350
// MI455X (gfx1250) — compile-verified
//
#include <hip/hip_runtime.h>
#include <hip/hip_bf16.h>

// ---------------- problem constants (from reference) ----------------
#define C   128
#define NU  100000
#define NI  120000
#define NE  800000
#define SS  4096
#define BB  8192
#define KK  20

typedef float v2f __attribute__((ext_vector_type(2)));
typedef float v8f __attribute__((ext_vector_type(8)));

#define FLAG_ACCUM 1
#define FLAG_RELU  2

// =====================================================================
// Generic NT GEMM on WMMA f32 16x16x4, 32x64 output tile per wave:
//   D[r][n] = sum_k A[arow(r)][k] * Bw[n][k]  (+ bias[n]) (+ r1row[r]*r1col[n*r1cs])
//   optional accumulate into existing D, relu, per-row scale.
// arow(r) = arowIdx ? arowIdx[r] : r / arowDiv.
// KD is compile-time (128 everywhere) -> fully unrolled reduction, all
// loads become base+immediate-offset, maximal load/WMMA overlap.
// Per k-step: 2 A-frag + 4 B-frag b64 loads feed 8 WMMAs (~10.7 FLOP/B vs L2).
// Requires M % 32 == 0, N % 64 == 0 at call sites (guards kept, but the
// WMMA stream itself is EXEC-uniform).
// Block = 32 threads (1 wave); grid = (M/32, N/64).
// =====================================================================
template <int KD>
__global__ void gemm_nt_wmma_f32(
    const float* __restrict__ A, int lda,
    const int* __restrict__ arowIdx, int arowDiv,
    const float* __restrict__ Bw, int ldb,
    float* __restrict__ D, int ldd,
    const float* __restrict__ bias,
    const float* __restrict__ r1row,
    const float* __restrict__ r1col, int r1cs,
    const float* __restrict__ rowscale,
    int M, int N, int flags)
{
  const int lane = threadIdx.x & 31;
  const int half = lane >> 4;      // 0: K=k,k+1   1: K=k+2,k+3
  const int l16  = lane & 15;
  const int m0 = blockIdx.x * 32;
  const int n0 = blockIdx.y * 64;
  if (m0 >= M || n0 >= N) return;  // wave-uniform exit

  // A fragments: lane%16 = M row within each 16-row tile
  const float* Ap[2];
#pragma unroll
  for (int mt = 0; mt < 2; ++mt) {
    int mr = m0 + mt * 16 + l16;
    if (mr > M - 1) mr = M - 1;                  // clamp (never hit at our call sites)
    long ar = arowIdx ? (long)arowIdx[mr] : (long)(mr / arowDiv);
    Ap[mt] = A + ar * (long)lda;
  }

  const float* Bp[4];
#pragma unroll
  for (int nt = 0; nt < 4; ++nt) {
    int nc = n0 + nt * 16 + l16;
    if (nc > N - 1) nc = N - 1;                  // clamp (never hit at our call sites)
    Bp[nt] = Bw + (long)nc * (long)ldb;
  }

  v8f acc[2][4];
#pragma unroll
  for (int mt = 0; mt < 2; ++mt)
#pragma unroll
    for (int nt = 0; nt < 4; ++nt)
      acc[mt][nt] = (v8f){0.f,0.f,0.f,0.f,0.f,0.f,0.f,0.f};

  if (flags & FLAG_ACCUM) {
#pragma unroll
    for (int mt = 0; mt < 2; ++mt)
#pragma unroll
      for (int nt = 0; nt < 4; ++nt) {
        const int nc = n0 + nt * 16 + l16;
#pragma unroll
        for (int v = 0; v < 8; ++v) {
          int mr = m0 + mt * 16 + v + 8 * half;  // C/D layout: VGPR v -> row v (+8 upper half)
          if (mr < M && nc < N) acc[mt][nt][v] = D[(long)mr * ldd + nc];
        }
      }
  }

#pragma unroll
  for (int k = 0; k < KD; k += 4) {
    const int ka = k + 2 * half;
    v2f a[2];
#pragma unroll
    for (int mt = 0; mt < 2; ++mt) {
      a[mt][0] = Ap[mt][ka];
      a[mt][1] = Ap[mt][ka + 1];
    }
#pragma unroll
    for (int nt = 0; nt < 4; ++nt) {
      v2f b;
      b[0] = Bp[nt][ka];
      b[1] = Bp[nt][ka + 1];
#pragma unroll
      for (int mt = 0; mt < 2; ++mt) {
        acc[mt][nt] = __builtin_amdgcn_wmma_f32_16x16x4_f32(
            /*neg_a=*/false, a[mt], /*neg_b=*/false, b,
            /*c_mod=*/(short)0, acc[mt][nt], /*reuse_a=*/false, /*reuse_b=*/false);
      }
    }
  }

#pragma unroll
  for (int nt = 0; nt < 4; ++nt) {
    const int nc = n0 + nt * 16 + l16;
    if (nc >= N) continue;
    const float bv = bias  ? bias[nc] : 0.f;
    const float cv = r1col ? r1col[(long)nc * r1cs] : 0.f;
#pragma unroll
    for (int mt = 0; mt < 2; ++mt) {
#pragma unroll
      for (int v = 0; v < 8; ++v) {
        int mr = m0 + mt * 16 + v + 8 * half;
        if (mr >= M) continue;
        float x = acc[mt][nt][v] + bv;
        if (r1row) x += r1row[mr] * cv;
        if (flags & FLAG_RELU) x = fmaxf(x, 0.f);
        if (rowscale) x *= rowscale[mr];
        D[(long)mr * ldd + nc] = x;
      }
    }
  }
}

// =====================================================================
// Top-K(20) + masked softmax per seed row of sim (SS x BB).
// One thread per seed; matches jax.lax.top_k tie-breaking (first index wins).
// =====================================================================
__global__ void topk_softmax_k(const float* __restrict__ sim,
                               const int* __restrict__ seed_t,
                               const int* __restrict__ bank_t,
                               float* __restrict__ sc,
                               int* __restrict__ idx)
{
  int s = blockIdx.x * blockDim.x + threadIdx.x;
  if (s >= SS) return;
  float vals[KK];
  int   ids[KK];
#pragma unroll
  for (int j = 0; j < KK; ++j) { vals[j] = -__builtin_inff(); ids[j] = 0; }
  const int st = seed_t[s];
  const float* row = sim + (long)s * BB;
  for (int b = 0; b < BB; ++b) {
    float v = (st < bank_t[b]) ? row[b] : -__builtin_inff();
    if (v > vals[KK - 1]) {
      int p = KK - 1;
      while (p > 0 && v > vals[p - 1]) {
        vals[p] = vals[p - 1]; ids[p] = ids[p - 1]; --p;
      }
      vals[p] = v; ids[p] = b;
    }
  }
  const float mx = vals[0];
  if (mx == -__builtin_inff()) {
    // all candidates invalid -> softmax is NaN -> nan_to_num -> 0
    for (int j = 0; j < KK; ++j) { sc[(long)s * KK + j] = 0.f; idx[(long)s * KK + j] = ids[j]; }
  } else {
    float e[KK], sum = 0.f;
#pragma unroll
    for (int j = 0; j < KK; ++j) { e[j] = __expf(vals[j] - mx); sum += e[j]; }
    const float inv = 1.f / sum;
#pragma unroll
    for (int j = 0; j < KK; ++j) {
      sc[(long)s * KK + j] = e[j] * inv;
      idx[(long)s * KK + j] = ids[j];
    }
  }
}

__global__ void gather_y_k(const float* __restrict__ bank_y,
                           const int* __restrict__ idx,
                           float* __restrict__ yflat, int n)
{
  int t = blockIdx.x * blockDim.x + threadIdx.x;
  if (t < n) yflat[t] = bank_y[idx[t]];
}

// h_agg[s][c] = sum_j msg[(s*K+j)][c]   (sc already applied in GEMM epilogue)
__global__ void agg_msgs_k(const float* __restrict__ msg, float* __restrict__ hagg)
{
  int s = blockIdx.x, c = threadIdx.x;
  float acc = 0.f;
#pragma unroll
  for (int j = 0; j < KK; ++j) acc += msg[((long)s * KK + j) * C + c];
  hagg[(long)s * C + c] = acc;
}

// ------------------- segment mean via float atomics -------------------
__global__ void edge_count_k(const int* __restrict__ dst, float* __restrict__ cnt, int n)
{
  int e = blockIdx.x * blockDim.x + threadIdx.x;
  if (e < n) atomicAdd(&cnt[dst[e]], 1.f);
}

__global__ void edge_scatter_k(const int* __restrict__ src, const int* __restrict__ dst,
                               const float* __restrict__ x, float* __restrict__ agg, long total)
{
  long t = (long)blockIdx.x * blockDim.x + threadIdx.x;
  if (t >= total) return;
  int e = (int)(t >> 7);       // C == 128
  int c = (int)(t & 127);
  atomicAdd(&agg[(long)dst[e] * C + c], x[(long)src[e] * C + c]);
}

__global__ void seg_div_k(float* __restrict__ agg, const float* __restrict__ cnt, long total)
{
  long t = (long)blockIdx.x * blockDim.x + threadIdx.x;
  if (t >= total) return;
  int i = (int)(t >> 7);
  agg[t] = agg[t] / fmaxf(cnt[i], 1.f);
}

// ------------------- LayerNorm + ReLU, one wave32 per row -------------------
__global__ void ln_relu_k(const float* __restrict__ src,
                          const float* __restrict__ g,
                          const float* __restrict__ b,
                          float* __restrict__ dst, int nrows)
{
  const int wave = threadIdx.x >> 5, lane = threadIdx.x & 31;
  const int row = blockIdx.x * 4 + wave;
  if (row >= nrows) return;
  const float* x = src + (long)row * C;
  float v[4], s = 0.f;
#pragma unroll
  for (int i = 0; i < 4; ++i) { v[i] = x[lane + 32 * i]; s += v[i]; }
#pragma unroll
  for (int o = 16; o > 0; o >>= 1) s += __shfl_xor(s, o, 32);
  const float mu = s * (1.f / C);
  float var = 0.f;
#pragma unroll
  for (int i = 0; i < 4; ++i) { float d = v[i] - mu; var += d * d; }
#pragma unroll
  for (int o = 16; o > 0; o >>= 1) var += __shfl_xor(var, o, 32);
  var *= (1.f / C);
  const float r = rsqrtf(var + 1e-5f);
#pragma unroll
  for (int i = 0; i < 4; ++i) {
    int c = lane + 32 * i;
    float y = (v[i] - mu) * r * g[c] + b[c];
    dst[(long)row * C + c] = fmaxf(y, 0.f);
  }
}

// =====================================================================
// Host orchestration.
// Input order (setup_inputs insertion order; 'params' expanded via JAX
// pytree flatten = dict keys sorted alphabetically):
//  0 x_user  1 x_item  2 bank_x_user  3 bank_y
//  4.. params: per layer (22 leaves):
//    conv.iu.{Wl,Wr,bl}, conv.ui.{Wl,Wr,bl},
//    ln.item.{b,g}, ln.user.{b,g},
//    sj.{Wk,Wq,bk,bq,mW1,mW2,mb1,mb2,uW1,uW2,ub1,ub2}
//  48 edge_src_ui  49 edge_dst_ui  50 edge_src_iu  51 edge_dst_iu
//  52 seed_time    53 bank_seed_time
// =====================================================================
extern "C" void kernel_launch(void* const* d_in, const int* in_sizes, int n_in,
                              void* d_out, int out_size, void* d_ws, size_t ws_size,
                              hipStream_t stream)
{
  (void)in_sizes; (void)n_in; (void)out_size; (void)ws_size;
  const float* in_xu  = (const float*)d_in[0];
  const float* in_xi  = (const float*)d_in[1];
  const float* bankx  = (const float*)d_in[2];
  const float* banky  = (const float*)d_in[3];
  const int PBASE = 4, PPL = 22;
  const int* e_src_ui = (const int*)d_in[PBASE + 2 * PPL + 0];
  const int* e_dst_ui = (const int*)d_in[PBASE + 2 * PPL + 1];
  const int* e_src_iu = (const int*)d_in[PBASE + 2 * PPL + 2];
  const int* e_dst_iu = (const int*)d_in[PBASE + 2 * PPL + 3];
  const int* seed_t   = (const int*)d_in[PBASE + 2 * PPL + 4];
  const int* bank_t   = (const int*)d_in[PBASE + 2 * PPL + 5];

  // ---- workspace carve (floats) ----
  float* wp = (float*)d_ws;
  auto take = [&](size_t n) { float* r = wp; wp += n; return r; };
  float* xu    = take((size_t)NU * C);
  float* xi    = take((size_t)NI * C);
  float* feat  = take((size_t)SS * C);
  float* kbuf  = take((size_t)SS * C);
  float* qbuf  = take((size_t)BB * C);
  float* hagg  = take((size_t)SS * C);
  float* scb   = take((size_t)SS * KK);
  int*   idxb  = (int*)take((size_t)SS * KK);
  float* yflat = take((size_t)SS * KK);
  float* cnt   = take((size_t)NI);
  float* scratch = wp;  // >= SS*BB floats; time-shared: sim | (hidM,msg) | hidU | (agg,out)

  hipMemcpyAsync(xu, in_xu, (size_t)NU * C * sizeof(float), hipMemcpyDeviceToDevice, stream);
  hipMemcpyAsync(xi, in_xi, (size_t)NI * C * sizeof(float), hipMemcpyDeviceToDevice, stream);

  auto gemm = [&](const float* A, int lda, const int* aIdx, int aDiv,
                  const float* Bw, int ldb, float* D, int ldd,
                  const float* bias, const float* r1row, const float* r1col, int r1cs,
                  const float* rowscale, int M, int N, int flags) {
    dim3 grid((unsigned)((M + 31) / 32), (unsigned)((N + 63) / 64));
    gemm_nt_wmma_f32<C><<<grid, 32, 0, stream>>>(A, lda, aIdx, aDiv, Bw, ldb, D, ldd,
                                                 bias, r1row, r1col, r1cs, rowscale,
                                                 M, N, flags);
  };

  for (int l = 0; l < 2; ++l) {
    auto PF = [&](int i) { return (const float*)d_in[PBASE + l * PPL + i]; };
    const float *iu_Wl = PF(0), *iu_Wr = PF(1), *iu_bl = PF(2);
    const float *ui_Wl = PF(3), *ui_Wr = PF(4), *ui_bl = PF(5);
    const float *ln_i_b = PF(6), *ln_i_g = PF(7), *ln_u_b = PF(8), *ln_u_g = PF(9);
    const float *Wk = PF(10), *Wq = PF(11), *bk = PF(12), *bq = PF(13);
    const float *mW1 = PF(14), *mW2 = PF(15), *mb1 = PF(16), *mb2 = PF(17);
    const float *uW1 = PF(18), *uW2 = PF(19), *ub1 = PF(20), *ub2 = PF(21);

    // ---------------- self join on first SS users ----------------
    hipMemcpyAsync(feat, xu, (size_t)SS * C * sizeof(float), hipMemcpyDeviceToDevice, stream);
    gemm(bankx, C, nullptr, 1, Wq, C, qbuf, C, bq, nullptr, nullptr, 0, nullptr, BB, C, 0);
    gemm(feat,  C, nullptr, 1, Wk, C, kbuf, C, bk, nullptr, nullptr, 0, nullptr, SS, C, 0);
    float* sim = scratch;
    gemm(kbuf, C, nullptr, 1, qbuf, C, sim, BB, nullptr, nullptr, nullptr, 0, nullptr, SS, BB, 0);
    topk_softmax_k<<<(SS + 127) / 128, 128, 0, stream>>>(sim, seed_t, bank_t, scb, idxb);
    gather_y_k<<<(SS * KK + 255) / 256, 256, 0, stream>>>(banky, idxb, yflat, SS * KK);

    // message MLP over SS*KK rows; mW1 is 128 x 257:
    //   hidden = h_i@mW1[:,0:128]^T + h_j@mW1[:,128:256]^T + y*mW1[:,256] + mb1, relu
    float* hidM = scratch;                       // sim is dead after topk
    float* msg  = scratch + (size_t)SS * KK * C;
    gemm(feat,  C, nullptr, KK,  mW1,         2 * C + 1, hidM, C, mb1, nullptr, nullptr, 0,
         nullptr, SS * KK, C, 0);
    gemm(bankx, C, idxb,    1,   mW1 + C,     2 * C + 1, hidM, C, nullptr, yflat, mW1 + 2 * C,
         2 * C + 1, nullptr, SS * KK, C, FLAG_ACCUM | FLAG_RELU);
    // msg = (hidden@mW2^T + mb2) * sc  (row-scale epilogue)
    gemm(hidM, C, nullptr, 1, mW2, C, msg, C, mb2, nullptr, nullptr, 0, scb, SS * KK, C, 0);
    agg_msgs_k<<<SS, C, 0, stream>>>(msg, hagg);

    // update MLP: feat_out = feat + relu([feat|hagg]@uW1^T + ub1)@uW2^T + ub2
    float* hidU = scratch;                       // hidM/msg dead
    gemm(feat, C, nullptr, 1, uW1,     2 * C, hidU, C, ub1, nullptr, nullptr, 0, nullptr, SS, C, 0);
    gemm(hagg, C, nullptr, 1, uW1 + C, 2 * C, hidU, C, nullptr, nullptr, nullptr, 0, nullptr,
         SS, C, FLAG_ACCUM | FLAG_RELU);
    // xu[:SS] currently == feat  ->  residual via accumulate
    gemm(hidU, C, nullptr, 1, uW2, C, xu, C, ub2, nullptr, nullptr, 0, nullptr, SS, C, FLAG_ACCUM);

    // ---------------- conv: user -> item ----------------
    float* agg  = scratch;                 // [0, NI*C)
    float* outb = scratch + (size_t)NI * C;
    hipMemsetAsync(agg, 0, (size_t)NI * C * sizeof(float), stream);
    hipMemsetAsync(cnt, 0, (size_t)NI * sizeof(float), stream);
    edge_count_k<<<(NE + 255) / 256, 256, 0, stream>>>(e_dst_ui, cnt, NE);
    edge_scatter_k<<<(unsigned)(((long)NE * C + 255) / 256), 256, 0, stream>>>(
        e_src_ui, e_dst_ui, xu, agg, (long)NE * C);
    seg_div_k<<<(unsigned)(((long)NI * C + 255) / 256), 256, 0, stream>>>(agg, cnt, (long)NI * C);
    gemm(agg, C, nullptr, 1, ui_Wl, C, outb, C, ui_bl, nullptr, nullptr, 0, nullptr, NI, C, 0);
    gemm(xi,  C, nullptr, 1, ui_Wr, C, outb, C, nullptr, nullptr, nullptr, 0, nullptr, NI, C, FLAG_ACCUM);

    // ---------------- conv: item -> user (needs OLD xi) ----------------
    hipMemsetAsync(agg, 0, (size_t)NU * C * sizeof(float), stream);
    hipMemsetAsync(cnt, 0, (size_t)NU * sizeof(float), stream);
    edge_count_k<<<(NE + 255) / 256, 256, 0, stream>>>(e_dst_iu, cnt, NE);
    edge_scatter_k<<<(unsigned)(((long)NE * C + 255) / 256), 256, 0, stream>>>(
        e_src_iu, e_dst_iu, xi, agg, (long)NE * C);
    seg_div_k<<<(unsigned)(((long)NU * C + 255) / 256), 256, 0, stream>>>(agg, cnt, (long)NU * C);

    // xi no longer needed as input -> LN item now (frees outb afterwards)
    ln_relu_k<<<(NI + 3) / 4, 128, 0, stream>>>(outb, ln_i_g, ln_i_b, xi, NI);

    gemm(agg, C, nullptr, 1, iu_Wl, C, outb, C, iu_bl, nullptr, nullptr, 0, nullptr, NU, C, 0);
    gemm(xu,  C, nullptr, 1, iu_Wr, C, outb, C, nullptr, nullptr, nullptr, 0, nullptr, NU, C, FLAG_ACCUM);
    ln_relu_k<<<(NU + 3) / 4, 128, 0, stream>>>(outb, ln_u_g, ln_u_b, xu, NU);
  }

  hipMemcpyAsync(d_out, xu, (size_t)NU * C * sizeof(float), hipMemcpyDeviceToDevice, stream);
  hipMemcpyAsync((float*)d_out + (size_t)NU * C, xi, (size_t)NI * C * sizeof(float),
                 hipMemcpyDeviceToDevice, stream);
}